// Arap_46059229282930
// MI455X (gfx1250) — compile-verified
//
#include <hip/hip_runtime.h>

typedef __attribute__((ext_vector_type(2))) float v2f;
typedef __attribute__((ext_vector_type(8))) float v8f;

#define WPB 8  // waves per block (wave32 -> 256 threads)

// Fast hardware approximations (v_rcp_f32 / v_rsq_f32 / v_sqrt_f32, ~1 ulp):
// plenty for a self-correcting Jacobi iteration, avoids IEEE div-scale/fixup chains.
__device__ __forceinline__ float frcp(float x)  { return __builtin_amdgcn_rcpf(x); }
__device__ __forceinline__ float frsq(float x)  { return __builtin_amdgcn_rsqf(x); }
__device__ __forceinline__ float fsqrt(float x) { return __builtin_amdgcn_sqrtf(x); }

// One two-sided Jacobi rotation on symmetric T, accumulating eigenvectors in Q (columns).
template<int P, int Qc>
__device__ __forceinline__ void jacobi_rot(float T[3][3], float Q[3][3]) {
  float apq = T[P][Qc];
  float c = 1.f, s = 0.f;
  if (__builtin_fabsf(apq) > 1e-20f) {
    float tau = (T[Qc][Qc] - T[P][P]) * 0.5f * frcp(apq);
    float t = (tau >= 0.f ? 1.f : -1.f) *
              frcp(__builtin_fabsf(tau) + fsqrt(1.f + tau * tau));
    c = frsq(1.f + t * t);
    s = t * c;
  }
#pragma unroll
  for (int k = 0; k < 3; ++k) { float a = T[k][P], b = T[k][Qc]; T[k][P] = c*a - s*b; T[k][Qc] = s*a + c*b; }
#pragma unroll
  for (int k = 0; k < 3; ++k) { float a = T[P][k], b = T[Qc][k]; T[P][k] = c*a - s*b; T[Qc][k] = s*a + c*b; }
#pragma unroll
  for (int k = 0; k < 3; ++k) { float a = Q[k][P], b = Q[k][Qc]; Q[k][P] = c*a - s*b; Q[k][Qc] = s*a + c*b; }
}

__global__ __launch_bounds__(256) void arap_fwd_kernel(
    const float* __restrict__ xyz1, const float* __restrict__ xyz2,
    const int* __restrict__ nbrList, const int* __restrict__ numNbr,
    const int* __restrict__ accNbr, const float* __restrict__ wMat,
    const float* __restrict__ arapW,
    float* __restrict__ outE, float* __restrict__ outR, int N) {
  // Per-edge slab (12 floats, 48B => float4-aligned rows):
  //   [0..2]=w*d1, [3]=0 | [4..6]=d2, [7]=0 | [8..10]=d1, [11]=0
  // Rows m>=3 of WMMA operands read the zero pads -> branchless ds_load.
  __shared__ __align__(16) float ed[WPB][16][12];

  const int tid  = threadIdx.x;
  const int wave = tid >> 5;
  const int lane = tid & 31;

  const int  b    = blockIdx.y;
  const int  v0   = blockIdx.x * WPB + wave;
  const bool live = v0 < N;
  const int  v    = live ? v0 : (N - 1);  // clamp: keep barriers/WMMA EXEC uniform

  const int start = accNbr[v];
  int count = numNbr[v];
  if (count > 16) count = 16;
  const int cm1 = (count > 0) ? (count - 1) : 0;

  const int   el     = lane & 15;
  const bool  valid  = el < count;
  const float validf = valid ? 1.f : 0.f;
  const int   e      = start + (el < cm1 ? el : cm1);  // clamped: load unconditionally
  const int   nbr    = nbrList[e];

  const float  aw    = arapW[0];
  const size_t vbase = (size_t)b * (size_t)N;

  float w = 0.f;
  if (lane < 16) {
    // rest-pose displacement + weight (lower half lanes)
    const float* c = xyz1 + (vbase + (size_t)v) * 3;
    const float* n = xyz1 + (vbase + (size_t)nbr) * 3;
    float d0 = (c[0] - n[0]) * validf;
    float d1 = (c[1] - n[1]) * validf;
    float d2 = (c[2] - n[2]) * validf;
    w = wMat[e] * validf;
    *(float4*)&ed[wave][el][0] = make_float4(w * d0, w * d1, w * d2, 0.f);
    *(float4*)&ed[wave][el][8] = make_float4(d0, d1, d2, 0.f);
  } else {
    // deformed-pose displacement (upper half lanes load xyz2 in parallel)
    const float* c = xyz2 + (vbase + (size_t)v) * 3;
    const float* n = xyz2 + (vbase + (size_t)nbr) * 3;
    *(float4*)&ed[wave][el][4] = make_float4((c[0] - n[0]) * validf,
                                             (c[1] - n[1]) * validf,
                                             (c[2] - n[2]) * validf, 0.f);
  }
  __syncthreads();

  // ---- Covariance S(3x3) = Wd1(3x16) * D2(16x3): 4 chained f32 WMMAs (K=4) ----
  // A 16x4 layout: lane m: v0=A[m][k0], v1=A[m][k0+1], k0 = 2*(lane>=16)
  // B 4x16 layout: lane n: v0=B[k0][n], v1=B[k0+1][n]
  const int m    = el;                 // WMMA row (also output/col index n)
  const int half = lane >> 4;
  const int k0   = half * 2;
  const int ma   = (m < 3) ? m : 3;    // component index; 3 => zero pad slot

  v2f Ac[4], Bc[4];
#pragma unroll
  for (int c4 = 0; c4 < 4; ++c4) {     // batch all 16 DS loads before the WMMAs
    const int e0 = 4 * c4 + k0;
    const int e1 = e0 + 1;
    Ac[c4].x = ed[wave][e0][ma];       // w*d1[m] of edge e0  (0 for m>=3)
    Ac[c4].y = ed[wave][e1][ma];
    Bc[c4].x = ed[wave][e0][4 + ma];   // d2[n] of edge e0    (0 for n>=3)
    Bc[c4].y = ed[wave][e1][4 + ma];
  }
  v8f acc = {};
#pragma unroll
  for (int c4 = 0; c4 < 4; ++c4)
    acc = __builtin_amdgcn_wmma_f32_16x16x4_f32(false, Ac[c4], false, Bc[c4],
                                                (short)0, acc, false, false);

  // Broadcast S[i][j] (VGPR i, lane j) to all lanes
  float S[3][3];
#pragma unroll
  for (int i = 0; i < 3; ++i) {
    float ci = acc[i];
    S[i][0] = __shfl(ci, 0);
    S[i][1] = __shfl(ci, 1);
    S[i][2] = __shfl(ci, 2);
  }

  const float detS = S[0][0]*(S[1][1]*S[2][2] - S[1][2]*S[2][1])
                   - S[0][1]*(S[1][0]*S[2][2] - S[1][2]*S[2][0])
                   + S[0][2]*(S[1][0]*S[2][1] - S[1][1]*S[2][0]);

  // ---- det-corrected polar of S^T : R = V U^T (reference rotation) ----
  // T = S S^T = U Sigma^2 U^T ; Jacobi eigendecompose -> Q (=U), lambda.
  float T[3][3];
#pragma unroll
  for (int i = 0; i < 3; ++i)
#pragma unroll
    for (int j = 0; j < 3; ++j)
      T[i][j] = S[i][0]*S[j][0] + S[i][1]*S[j][1] + S[i][2]*S[j][2];

  float Q[3][3] = {{1.f,0.f,0.f},{0.f,1.f,0.f},{0.f,0.f,1.f}};
#pragma unroll
  for (int sweep = 0; sweep < 5; ++sweep) {
    jacobi_rot<0,1>(T, Q);
    jacobi_rot<0,2>(T, Q);
    jacobi_rot<1,2>(T, Q);
  }

  // flip column of smallest singular value when det(S) < 0
  int imin = 0; float lmin = T[0][0];
  if (T[1][1] < lmin) { imin = 1; lmin = T[1][1]; }
  if (T[2][2] < lmin) { imin = 2; }
  const float neg = (detS < 0.f) ? -1.f : 1.f;
  const float sg[3] = { (imin == 0) ? neg : 1.f,
                        (imin == 1) ? neg : 1.f,
                        (imin == 2) ? neg : 1.f };

  // R = sum_k (sg_k/sigma_k) (S^T q_k) q_k^T
  float R[3][3] = {};
#pragma unroll
  for (int k = 0; k < 3; ++k) {
    float lam = T[k][k] > 0.f ? T[k][k] : 0.f;
    float sigma = fsqrt(lam);
    float inv = sg[k] * frcp(sigma > 1e-12f ? sigma : 1e-12f);
    float u0 = (S[0][0]*Q[0][k] + S[1][0]*Q[1][k] + S[2][0]*Q[2][k]) * inv;
    float u1 = (S[0][1]*Q[0][k] + S[1][1]*Q[1][k] + S[2][1]*Q[2][k]) * inv;
    float u2 = (S[0][2]*Q[0][k] + S[1][2]*Q[1][k] + S[2][2]*Q[2][k]) * inv;
#pragma unroll
    for (int j = 0; j < 3; ++j) {
      R[0][j] += u0 * Q[j][k];
      R[1][j] += u1 * Q[j][k];
      R[2][j] += u2 * Q[j][k];
    }
  }

  // ---- Rd1(3x16) = R(3x3, K-padded) * D1(3x16) in one WMMA ----
  // A from registers via cndmask chains (no LDS, no exec games)
  const float rc0 = (m == 0) ? R[0][0] : (m == 1) ? R[1][0] : (m == 2) ? R[2][0] : 0.f;
  const float rc1 = (m == 0) ? R[0][1] : (m == 1) ? R[1][1] : (m == 2) ? R[2][1] : 0.f;
  const float rc2 = (m == 0) ? R[0][2] : (m == 1) ? R[1][2] : (m == 2) ? R[2][2] : 0.f;
  v2f Ar, Br;
  Ar.x = half ? rc2 : rc0;                 // A[m][k0], k0 in {0,2}
  Ar.y = half ? 0.f : rc1;                 // A[m][k0+1], k=3 -> 0
  Br.x = ed[wave][el][8 + k0];             // d1[k0] of edge n (=el)
  Br.y = ed[wave][el][8 + (half ? 3 : 1)]; // d1[1] or zero pad
  v8f zero = {};
  v8f rd = __builtin_amdgcn_wmma_f32_16x16x4_f32(false, Ar, false, Br,
                                                 (short)0, zero, false, false);

  // ---- residual + energy reduction: lane e holds (R d1_e)[i] in rd[i] ----
  float r0 = 0.f, r1 = 0.f, r2 = 0.f;
  if (lane < 16) {
    float t0 = ed[wave][el][4] - rd[0];
    float t1 = ed[wave][el][5] - rd[1];
    float t2 = ed[wave][el][6] - rd[2];
    r0 = w * t0 * t0; r1 = w * t1 * t1; r2 = w * t2 * t2;
  }
#pragma unroll
  for (int off = 1; off < 16; off <<= 1) {
    r0 += __shfl_xor(r0, off);
    r1 += __shfl_xor(r1, off);
    r2 += __shfl_xor(r2, off);
  }

  if (live && lane == 0) {
    const size_t base = vbase + (size_t)v;
    outE[base * 3 + 0] = aw * r0;
    outE[base * 3 + 1] = aw * r1;
    outE[base * 3 + 2] = aw * r2;
    float* orow = outR + base * 9;
    orow[0] = R[0][0]; orow[1] = R[0][1]; orow[2] = R[0][2];
    orow[3] = R[1][0]; orow[4] = R[1][1]; orow[5] = R[1][2];
    orow[6] = R[2][0]; orow[7] = R[2][1]; orow[8] = R[2][2];
  }
}

extern "C" void kernel_launch(void* const* d_in, const int* in_sizes, int n_in,
                              void* d_out, int out_size, void* d_ws, size_t ws_size,
                              hipStream_t stream) {
  const float* xyz1  = (const float*)d_in[0];
  const float* xyz2  = (const float*)d_in[1];
  const int*   nbrL  = (const int*)d_in[2];
  const int*   numN  = (const int*)d_in[3];
  const int*   accN  = (const int*)d_in[4];
  const float* wMat  = (const float*)d_in[5];
  const float* arapW = (const float*)d_in[6];

  const int N = in_sizes[3];               // numNeighbors has N entries
  const int B = in_sizes[0] / (3 * N);     // xyz1 is B*N*3
  const int total = B * N;

  float* outE = (float*)d_out;             // [B,N,3]
  float* outR = outE + (size_t)total * 3;  // [B,N,9]

  dim3 grid((N + WPB - 1) / WPB, B, 1);    // one wave32 per (batch, vertex)
  arap_fwd_kernel<<<grid, WPB * 32, 0, stream>>>(
      xyz1, xyz2, nbrL, numN, accN, wMat, arapW, outE, outR, N);
}